// ROIPooling_63479616635497
// MI455X (gfx1250) — compile-verified
//
#include <hip/hip_runtime.h>
#include <float.h>

// Problem constants (from reference setup_inputs)
#define B_    4
#define C_    256
#define H_    50
#define W_    50
#define R_    300
#define PH_   7
#define PW_   7

#define CCHUNK   32          // channels staged per block
#define STRH_MAX 8           // max adaptive bin height for H=50,PH=7 (proven <=8)
#define WPAD     51          // roi_w <= 50, +1 pad to break LDS bank alignment
#define NTHREADS 256         // 8 wave32 waves
#define NWAVES   8
// LDS: 32 * 8 * 51 * 4 = 52224 B (~51 KB of the 320 KB/WGP pool)

// CDNA5 async global->LDS path (guarded; falls back to load+ds_store if absent)
#if defined(__gfx1250__) && __has_builtin(__builtin_amdgcn_global_load_async_to_lds_b32)
#define USE_ASYNC_LDS 1
// clang's parameter types for this builtin: (as1 int*, as3 int*, imm int, imm int)
typedef __attribute__((address_space(1))) int g_as1_int;
typedef __attribute__((address_space(3))) int l_as3_int;
#else
#define USE_ASYNC_LDS 0
#endif

__device__ __forceinline__ void wait_asynccnt0() {
#if defined(__gfx1250__)
#if __has_builtin(__builtin_amdgcn_s_wait_asynccnt)
    __builtin_amdgcn_s_wait_asynccnt(0);
#else
    asm volatile("s_wait_asynccnt 0" ::: "memory");
#endif
#endif
}

__global__ __launch_bounds__(NTHREADS) void roi_pool_gfx1250(
    const float* __restrict__ fm,     // [B,C,H,W]
    const float* __restrict__ rois,   // [R,5] = x0,y0,x1,y1,batch
    float* __restrict__ out)          // [R,C,PH,PW]
{
    __shared__ float tile[CCHUNK * STRH_MAX * WPAD];

    const int r    = blockIdx.x;          // ROI
    const int c0   = blockIdx.y * CCHUNK; // first channel of this chunk
    const int tid  = threadIdx.x;
    const int lane = tid & 31;            // wave32
    const int wave = tid >> 5;

    // ROI decode — faithful to reference: rintf (nearest-even == jnp.round),
    // clamp to [0,dim-1], then x1=max(x1,x0+1) / y1=max(y1,y0+1).
    const float rx0 = rois[r * 5 + 0];
    const float ry0 = rois[r * 5 + 1];
    const float rx1 = rois[r * 5 + 2];
    const float ry1 = rois[r * 5 + 3];
    const int   b   = (int)rois[r * 5 + 4];   // truncation, values in [0,B)

    int x0 = (int)rintf(rx0); x0 = min(max(x0, 0), W_ - 1);
    int x1 = (int)rintf(rx1); x1 = min(max(x1, 0), W_ - 1);
    int y0 = (int)rintf(ry0); y0 = min(max(y0, 0), H_ - 1);
    int y1 = (int)rintf(ry1); y1 = min(max(y1, 0), H_ - 1);
    x1 = max(x1, x0 + 1);   // may reach 50; all sampled cols stay <= 49
    y1 = max(y1, y0 + 1);
    const int rw = x1 - x0; // 1..50
    const int rh = y1 - y0; // 1..50

    const float* fmb = fm + ((size_t)b * C_ + c0) * (size_t)(H_ * W_) + x0;

    // Compute-phase assignment: threads 0..(CCHUNK*PW-1) -> (channel-in-chunk, pw)
    const int cl = tid / PW_;             // compile-time divisor -> cheap
    const int pw = tid - cl * PW_;
    int ws = 0, wcnt = 0;
    if (cl < CCHUNK) {
        ws   = (pw * rw) / PW_;                                // bin start (rel. x0)
        wcnt = ((pw + 1) * rw + PW_ - 1) / PW_ - ws;           // bin width, 1..8
    }

    for (int ph = 0; ph < PH_; ++ph) {
        const int hs = y0 + (ph * rh) / PH_;
        const int he = y0 + ((ph + 1) * rh + PH_ - 1) / PH_;   // <= y1 <= 50
        const int sh = he - hs;                                 // 1..8

        // Stage strip [hs,he) x [x0,x0+rw) for CCHUNK channels.
        // wave -> channel (stride NWAVES), lane -> column: no divides, coalesced rows.
        for (int c = wave; c < CCHUNK; c += NWAVES) {
            const float* gch = fmb + (size_t)c * (H_ * W_) + hs * W_;
            float*       lch = &tile[c * (STRH_MAX * WPAD)];
            for (int hh = 0; hh < sh; ++hh) {
                for (int ww = lane; ww < rw; ww += 32) {
#if USE_ASYNC_LDS
                    __builtin_amdgcn_global_load_async_to_lds_b32(
                        (g_as1_int*)(gch + hh * W_ + ww),
                        (l_as3_int*)(lch + hh * WPAD + ww), 0, 0);
#else
                    lch[hh * WPAD + ww] = gch[hh * W_ + ww];
#endif
                }
            }
        }
#if USE_ASYNC_LDS
        wait_asynccnt0();       // each wave drains its own ASYNCcnt before the barrier
#endif
        __syncthreads();

        // Reduce bins from LDS: 224 active threads, <= 64 fmax each.
        if (cl < CCHUNK) {
            float m = -FLT_MAX; // == jnp.finfo(f32).min; bins are provably nonempty
            const float* base = &tile[cl * (STRH_MAX * WPAD) + ws];
            for (int hh = 0; hh < sh; ++hh) {
                const float* rowp = base + hh * WPAD;
                for (int k = 0; k < wcnt; ++k)
                    m = fmaxf(m, rowp[k]);
            }
            // write-once output: nontemporal store, keep L2 for the reused fm
            __builtin_nontemporal_store(
                m, &out[(((size_t)r * C_ + (c0 + cl)) * PH_ + ph) * PW_ + pw]);
        }
        __syncthreads();        // tile reused next strip
    }
}

extern "C" void kernel_launch(void* const* d_in, const int* in_sizes, int n_in,
                              void* d_out, int out_size, void* d_ws, size_t ws_size,
                              hipStream_t stream) {
    const float* fm   = (const float*)d_in[0];   // 4*256*50*50 f32
    const float* rois = (const float*)d_in[1];   // 300*5 f32
    float*       out  = (float*)d_out;           // 300*256*7*7 f32

    dim3 grid(R_, C_ / CCHUNK, 1);               // 300 x 8 blocks
    roi_pool_gfx1250<<<grid, NTHREADS, 0, stream>>>(fm, rois, out);
}